// CRFModule_61538291417672
// MI455X (gfx1250) — compile-verified
//
#include <hip/hip_runtime.h>
#include <hip/hip_bf16.h>
#include <stdint.h>

#define B_TOT 1024
#define S_LEN 512
#define T_TAG 48
#define BGRP  16      // batches per workgroup (WMMA N dimension)
#define FPAD  52      // fbuf row stride in floats (16B-aligned rows, conflict-free reads)

#if defined(__has_builtin)
#  if __has_builtin(__builtin_amdgcn_global_load_async_to_lds_b128)
#    define HAVE_ASYNC_LDS 1
#  endif
#endif
#ifndef HAVE_ASYNC_LDS
#  define HAVE_ASYNC_LDS 0
#endif

typedef float v2f __attribute__((ext_vector_type(2)));
typedef float v8f __attribute__((ext_vector_type(8)));

// Raw v_log_f32 (log2): inputs here are bounded away from 0/denorm/inf,
// so skip OCML's subnormal-rescale + extended-precision + inf guards.
__device__ __forceinline__ float fast_log2(float x) {
#if defined(__has_builtin) && __has_builtin(__builtin_amdgcn_logf)
    return __builtin_amdgcn_logf(x);
#else
    return __log2f(x);
#endif
}

#if HAVE_ASYNC_LDS
typedef int v4i __attribute__((vector_size(4 * sizeof(int))));
#define GLOBAL_AS __attribute__((address_space(1)))
#define LDS_AS    __attribute__((address_space(3)))
// One feats row segment per thread: 32 B = 2x async b128 straight into LDS.
__device__ __forceinline__ void async_feats_load(const float* gsrc, float* ldst) {
    __builtin_amdgcn_global_load_async_to_lds_b128(
        (GLOBAL_AS v4i*)gsrc, (LDS_AS v4i*)ldst, 0, 0);
    __builtin_amdgcn_global_load_async_to_lds_b128(
        (GLOBAL_AS v4i*)(gsrc + 4), (LDS_AS v4i*)(ldst + 4), 0, 0);
}
__device__ __forceinline__ void wait_async_le2() {
#if __has_builtin(__builtin_amdgcn_s_wait_asynccnt)
    __builtin_amdgcn_s_wait_asynccnt(2);
#else
    asm volatile("s_wait_asynccnt 2" ::: "memory");
#endif
}
__device__ __forceinline__ void wait_async_0() {
#if __has_builtin(__builtin_amdgcn_s_wait_asynccnt)
    __builtin_amdgcn_s_wait_asynccnt(0);
#else
    asm volatile("s_wait_asynccnt 0" ::: "memory");
#endif
}
#endif

// One workgroup = 16 batches, 3 waves (wave w owns output-tag rows [16w, 16w+16)).
// fv is kept in WMMA C/D register layout across the whole time scan.
__global__ void __launch_bounds__(96)
crf_fwd_kernel(const float* __restrict__ feats,
               const int*   __restrict__ tags,
               const uint8_t* __restrict__ mask,
               const float* __restrict__ trans,
               const float* __restrict__ startt,
               const float* __restrict__ stopt,
               float* __restrict__ partial)
{
    __shared__ float trans_lds[T_TAG * T_TAG];            // 9216 B
    __shared__ __align__(16) float fbufs[2][BGRP * FPAD]; // 2x 3328 B (double-buffered feats tile)
    __shared__ v2f   ev2[24 * 16];                        // 3072 B, ev in B-operand layout
    __shared__ float pbuf[96];                            // per-lane partials (max / sum)

    const int tid  = threadIdx.x;
    const int wave = tid >> 5;
    const int lane = tid & 31;
    const int half = lane >> 4;
    const int l16  = lane & 15;
    const int mt   = wave;                 // M-tile index
    const int b0   = blockIdx.x * BGRP;

    // ---- transitions -> LDS, global max for scaling ----
    float pm = -3.0e38f;
    for (int k = tid; k < T_TAG * T_TAG; k += 96) {
        float v = trans[k];
        trans_lds[k] = v;
        pm = fmaxf(pm, v);
    }
    pbuf[tid] = pm;
    __syncthreads();
    float Mmax = -3.0e38f;
    for (int k = 0; k < 96; ++k) Mmax = fmaxf(Mmax, pbuf[k]);
    __syncthreads();

    // ---- constant A operand: P[i,j] = exp(M[i,j] - Mmax), WMMA A layout ----
    v2f a[12];
    {
        const int i = mt * 16 + l16;
        #pragma unroll
        for (int c = 0; c < 12; ++c) {
            const int j0 = 4 * c + 2 * half;
            a[c].x = __expf(trans_lds[i * T_TAG + j0    ] - Mmax);
            a[c].y = __expf(trans_lds[i * T_TAG + j0 + 1] - Mmax);
        }
    }

    // ---- cooperative feats loader: 96 threads cover 16 rows x 48 floats ----
    const int frow = tid / 6;              // 0..15 (batch within group)
    const int fcol = (tid % 6) * 8;        // 0,8,...,40
    const float* fptr = feats + (size_t)(b0 + frow) * S_LEN * T_TAG + fcol;
    const int foff = frow * FPAD + fcol;
    // fv-update read base: 8 consecutive floats, 16B aligned
    const int fread = l16 * FPAD + mt * 16 + 8 * half;

    // ---- t = 0 (and t = 1 prefetch) ----
#if HAVE_ASYNC_LDS
    async_feats_load(fptr,         &fbufs[0][foff]);   // feats(0)
    async_feats_load(fptr + T_TAG, &fbufs[1][foff]);   // feats(1)
    wait_async_le2();                                  // feats(0) resident
    __syncthreads();
#else
    {
        float4 fA = *(const float4*)(fptr);
        float4 fB = *(const float4*)(fptr + 4);
        *(float4*)&fbufs[0][foff]     = fA;
        *(float4*)&fbufs[0][foff + 4] = fB;
    }
    __syncthreads();
#endif

    float acc[8];                          // fv in C/D layout: i = mt*16 + v + 8*half, b = l16
    #pragma unroll
    for (int v = 0; v < 8; ++v) {
        const int i = mt * 16 + v + 8 * half;
        acc[v] = fbufs[0][l16 * FPAD + i] + startt[i];
    }

    // gold-path state (wave 0, lanes 0..15 each own one batch)
    float gold = 0.0f;
    int tagp = 0, lasttag = 0;
    if (wave == 0 && lane < 16) {
        int t0 = tags[(size_t)(b0 + l16) * S_LEN];
        gold = fbufs[0][l16 * FPAD + t0] + startt[t0];
        tagp = t0; lasttag = t0;
    }

    // ---- sequential scan ----
    for (int t = 1; t < S_LEN; ++t) {
#if HAVE_ASYNC_LDS
        const float* fbr = fbufs[t & 1];
#else
        const float* fbr = fbufs[0];
        const float* fp = fptr + (size_t)t * T_TAG;
        float4 fA = *(const float4*)(fp);
        float4 fB = *(const float4*)(fp + 4);
        if (t + 1 < S_LEN) __builtin_prefetch(fp + T_TAG, 0, 3);
#endif

        // per-batch max of fv: per-lane partial, combine 6-way via LDS
        float pmax = acc[0];
        #pragma unroll
        for (int v = 1; v < 8; ++v) pmax = fmaxf(pmax, acc[v]);
        pbuf[tid] = pmax;
        __syncthreads();                                   // barrier 1

#if HAVE_ASYNC_LDS
        // feats(t+1) -> other buffer; its last readers finished before barrier 1
        if (t + 1 < S_LEN)
            async_feats_load(fptr + (size_t)(t + 1) * T_TAG, &fbufs[(t + 1) & 1][foff]);
#endif

        float m = fmaxf(fmaxf(fmaxf(pbuf[l16],      pbuf[l16 + 16]),
                              fmaxf(pbuf[32 + l16], pbuf[32 + l16 + 16])),
                        fmaxf(pbuf[64 + l16], pbuf[64 + l16 + 16]));

        // ev = exp(fv - m) stored directly in WMMA B-operand layout (pairs along K)
        #pragma unroll
        for (int vp = 0; vp < 4; ++vp) {
            v2f e;
            e.x = __expf(acc[2 * vp]     - m);
            e.y = __expf(acc[2 * vp + 1] - m);
            const int p = mt * 8 + vp + 4 * half;          // j >> 1
            ev2[p * 16 + l16] = e;
        }

#if HAVE_ASYNC_LDS
        // retire the feats(t) pair (2 newest = feats(t+1) may stay in flight)
        if (t + 1 < S_LEN) wait_async_le2(); else wait_async_0();
#else
        *(float4*)&fbufs[0][foff]     = fA;
        *(float4*)&fbufs[0][foff + 4] = fB;
#endif
        __syncthreads();                                   // barrier 2

        // dot[i,b] = sum_j P[i,j] * ev[j,b] : 12 chained fp32 WMMAs
        v8f dot = {};
        #pragma unroll
        for (int c = 0; c < 12; ++c) {
            v2f bb = ev2[(2 * c + half) * 16 + l16];
            dot = __builtin_amdgcn_wmma_f32_16x16x4_f32(
                false, a[c], false, bb, (short)0, dot, false, false);
        }

        // preload this lane's 8 consecutive feats values: 2x ds_load_b128, one wait
        float4 f03 = *(const float4*)&fbr[fread];
        float4 f47 = *(const float4*)&fbr[fread + 4];
        float fvals[8] = { f03.x, f03.y, f03.z, f03.w, f47.x, f47.y, f47.z, f47.w };

        // fv' = feat + log(dot) + m + Mmax, branchless masked update
        const uint8_t mk = mask[(size_t)(b0 + l16) * S_LEN + t];
        const float mm = m + Mmax;
        #pragma unroll
        for (int v = 0; v < 8; ++v) {
            float nf = fmaf(0.69314718055994531f, fast_log2(dot[v]), fvals[v] + mm);
            asm volatile("" : "+v"(nf));   // keep select as v_cndmask, no exec branching
            acc[v] = mk ? nf : acc[v];
        }

        // gold path (reuses LDS feats tile + transitions), branchless
        if (wave == 0 && lane < 16) {
            const size_t bi = (size_t)(b0 + l16) * S_LEN + t;
            int tc = tags[bi];
            bool mb = mask[bi] != 0;
            float contrib = trans_lds[tc * T_TAG + tagp] + fbr[l16 * FPAD + tc];
            gold   += mb ? contrib : 0.0f;
            lasttag = mb ? tc : lasttag;
            tagp = tc;
        }
    }

    // ---- finalize: log_z = LSE_i(fv_i + stop_i) ----
    float sv[8];
    float pmax = -3.0e38f;
    #pragma unroll
    for (int v = 0; v < 8; ++v) {
        const int i = mt * 16 + v + 8 * half;
        sv[v] = acc[v] + stopt[i];
        pmax = fmaxf(pmax, sv[v]);
    }
    __syncthreads();
    pbuf[tid] = pmax;
    __syncthreads();
    float m2 = fmaxf(fmaxf(fmaxf(pbuf[l16],      pbuf[l16 + 16]),
                           fmaxf(pbuf[32 + l16], pbuf[32 + l16 + 16])),
                     fmaxf(pbuf[64 + l16], pbuf[64 + l16 + 16]));
    float ps = 0.0f;
    #pragma unroll
    for (int v = 0; v < 8; ++v) ps += __expf(sv[v] - m2);
    __syncthreads();
    pbuf[tid] = ps;
    __syncthreads();
    if (wave == 0 && lane < 16) {
        float s = pbuf[l16] + pbuf[l16 + 16] + pbuf[32 + l16] + pbuf[32 + l16 + 16]
                + pbuf[64 + l16] + pbuf[64 + l16 + 16];
        float logz = 0.69314718055994531f * fast_log2(s) + m2;
        gold += stopt[lasttag];
        partial[blockIdx.x * BGRP + l16] = logz - gold;
    }
}

// Deterministic fixed-order mean over the 1024 per-batch partials.
__global__ void crf_reduce_kernel(const float* __restrict__ partial,
                                  float* __restrict__ out)
{
    __shared__ float s[256];
    const int tid = threadIdx.x;
    float v = 0.0f;
    for (int k = tid; k < B_TOT; k += 256) v += partial[k];
    s[tid] = v;
    __syncthreads();
    for (int w = 128; w > 0; w >>= 1) {
        if (tid < w) s[tid] += s[tid + w];
        __syncthreads();
    }
    if (tid == 0) out[0] = s[0] / (float)B_TOT;
}

extern "C" void kernel_launch(void* const* d_in, const int* in_sizes, int n_in,
                              void* d_out, int out_size, void* d_ws, size_t ws_size,
                              hipStream_t stream) {
    const float*   feats  = (const float*)d_in[0];
    const int*     tags   = (const int*)d_in[1];
    const uint8_t* mask   = (const uint8_t*)d_in[2];
    const float*   trans  = (const float*)d_in[3];
    const float*   startt = (const float*)d_in[4];
    const float*   stopt  = (const float*)d_in[5];
    float* partial = (float*)d_ws;   // 1024 floats

    crf_fwd_kernel<<<B_TOT / BGRP, 96, 0, stream>>>(
        feats, tags, mask, trans, startt, stopt, partial);
    crf_reduce_kernel<<<1, 256, 0, stream>>>(partial, (float*)d_out);
}